// KnowledgeEncoder_7370163880628
// MI455X (gfx1250) — compile-verified
//
#include <hip/hip_runtime.h>
#include <math.h>

// ---- problem constants (match reference) ----
#define CB 16
#define CM 4096
#define CT 4
#define CD 256
#define CV 40000
#define CHOPS 3
#define CNLAYER 3
#define CCONV 512
#define CBD (CB*CD)                       // 4096
#define CBM (CB*CM)                       // 65536
#define CBMD ((long long)CB*(long long)CM*(long long)CD)  // 16,777,216

#define MODE_COPY 0
#define MODE_GATE 1
#define MODE_RELU 2

typedef __attribute__((ext_vector_type(2))) float v2f;
typedef __attribute__((ext_vector_type(8))) float v8f;

// ---------------------------------------------------------------------------
// Kernel 1: embedding gather.  One wave32 per (table h, b, m) row.
// E[h][b,m,:] = sum_t C_emb[h][story[b,m,t],:]  (+ history window add)
// h<3 -> workspace, h==3 -> m_story output slot directly.
// ---------------------------------------------------------------------------
__global__ void embed_kernel(const int* __restrict__ story,
                             const int* __restrict__ kb_len,
                             const int* __restrict__ conv_len,
                             const float* __restrict__ hist,
                             const float* __restrict__ C_emb,
                             float* __restrict__ Ews,
                             float* __restrict__ E3)
{
    int wid  = threadIdx.x >> 5;
    int lane = threadIdx.x & 31;
    int row  = blockIdx.x * 8 + wid;          // 0 .. 4*B*M-1
    int h    = row / CBM;
    int bm   = row - h * CBM;
    int b    = bm >> 12;                       // / M
    int m    = bm & (CM - 1);

    const int* st = story + (long long)bm * CT;
    const float* tab = C_emb + (long long)h * CV * CD;

    float4 a0 = make_float4(0.f, 0.f, 0.f, 0.f);
    float4 a1 = make_float4(0.f, 0.f, 0.f, 0.f);
#pragma unroll
    for (int t = 0; t < CT; ++t) {
        int idx = st[t];
        const float4* r = (const float4*)(tab + (long long)idx * CD) + lane * 2;
        float4 x0 = r[0], x1 = r[1];
        a0.x += x0.x; a0.y += x0.y; a0.z += x0.z; a0.w += x0.w;
        a1.x += x1.x; a1.y += x1.y; a1.z += x1.z; a1.w += x1.w;
    }
    int kb = kb_len[b], cl = conv_len[b];
    if (m >= kb && m < kb + cl) {
        const float4* hr = (const float4*)(hist + ((long long)b * CCONV + (m - kb)) * CD) + lane * 2;
        float4 h0 = hr[0], h1 = hr[1];
        a0.x += h0.x; a0.y += h0.y; a0.z += h0.z; a0.w += h0.w;
        a1.x += h1.x; a1.y += h1.y; a1.z += h1.z; a1.w += h1.w;
    }
    float* dst = (h < 3) ? (Ews + (long long)h * CBMD + (long long)bm * CD)
                         : (E3 + (long long)bm * CD);
    float4* dv = (float4*)dst + lane * 2;
    dv[0] = a0; dv[1] = a1;
}

// ---------------------------------------------------------------------------
// Kernel 2: batched GEMV  s[b,m] = dot(E[b,m,:], u[b,:]) (* gp[b,m] if given)
// One wave32 per row, lanes stride D, shuffle reduction.
// ---------------------------------------------------------------------------
__global__ void logits_kernel(const float* __restrict__ E,
                              const float* __restrict__ u,
                              const float* __restrict__ gp,
                              float* __restrict__ s)
{
    int wid  = threadIdx.x >> 5;
    int lane = threadIdx.x & 31;
    int row  = blockIdx.x * 8 + wid;          // b*M + m
    int b    = row >> 12;

    const float4* e  = (const float4*)(E + (long long)row * CD) + lane * 2;
    const float4* uv = (const float4*)(u + b * CD) + lane * 2;
    float4 p0 = e[0], p1 = e[1];
    float4 q0 = uv[0], q1 = uv[1];
    float dot = p0.x*q0.x + p0.y*q0.y + p0.z*q0.z + p0.w*q0.w
              + p1.x*q1.x + p1.y*q1.y + p1.z*q1.z + p1.w*q1.w;
#pragma unroll
    for (int off = 16; off >= 1; off >>= 1)
        dot += __shfl_xor(dot, off, 32);
    if (gp) dot *= gp[row];
    if (lane == 0) s[row] = dot;
}

// ---------------------------------------------------------------------------
// Kernel 3: per-batch softmax stats (max, sum of exp) + zero the o accumulator
// ---------------------------------------------------------------------------
__global__ void softmax_stats_kernel(const float* __restrict__ s,
                                     float* __restrict__ stats,
                                     float* __restrict__ o)
{
    __shared__ float red[256];
    int b = blockIdx.x, t = threadIdx.x;
    float mx = -INFINITY;
    for (int m = t; m < CM; m += 256) mx = fmaxf(mx, s[b * CM + m]);
    red[t] = mx; __syncthreads();
    for (int off = 128; off; off >>= 1) { if (t < off) red[t] = fmaxf(red[t], red[t + off]); __syncthreads(); }
    mx = red[0]; __syncthreads();
    float sum = 0.f;
    for (int m = t; m < CM; m += 256) sum += __expf(s[b * CM + m] - mx);
    red[t] = sum; __syncthreads();
    for (int off = 128; off; off >>= 1) { if (t < off) red[t] += red[t + off]; __syncthreads(); }
    if (t == 0) { stats[b * 2] = mx; stats[b * 2 + 1] = red[0]; }
    o[b * CD + t] = 0.f;
}

// ---------------------------------------------------------------------------
// Kernel 4: o[b,:] += sum_m softmax(s)[b,m] * (gp[b,m]) * E[b,m,:]
// grid = B*16 chunk blocks; wave covers all D; LDS accumulate then one
// global f32 atomic per element.  Optionally emits psoft to d_out.
// ---------------------------------------------------------------------------
__global__ void weighted_kernel(const float* __restrict__ E,
                                const float* __restrict__ s,
                                const float* __restrict__ stats,
                                const float* __restrict__ gp,
                                float* __restrict__ o,
                                float* __restrict__ psoft)
{
    __shared__ float acc[CD];
    int b     = blockIdx.x >> 4;
    int chunk = blockIdx.x & 15;
    int t     = threadIdx.x;
    acc[t] = 0.f;
    __syncthreads();

    int wid = t >> 5, lane = t & 31;
    float mx  = stats[b * 2];
    float inv = 1.0f / stats[b * 2 + 1];
    float4 A0 = make_float4(0.f, 0.f, 0.f, 0.f);
    float4 A1 = make_float4(0.f, 0.f, 0.f, 0.f);

    int m0 = chunk * 256 + wid * 32;
    for (int i = 0; i < 32; ++i) {
        int row = b * CM + m0 + i;
        float p = __expf(s[row] - mx) * inv;
        if (psoft && lane == 0) psoft[row] = p;
        float w = gp ? p * gp[row] : p;
        const float4* e = (const float4*)(E + (long long)row * CD) + lane * 2;
        float4 e0 = e[0], e1 = e[1];
        A0.x += e0.x * w; A0.y += e0.y * w; A0.z += e0.z * w; A0.w += e0.w * w;
        A1.x += e1.x * w; A1.y += e1.y * w; A1.z += e1.z * w; A1.w += e1.w * w;
    }
    float* ap = acc + lane * 8;
    atomicAdd(ap + 0, A0.x); atomicAdd(ap + 1, A0.y);
    atomicAdd(ap + 2, A0.z); atomicAdd(ap + 3, A0.w);
    atomicAdd(ap + 4, A1.x); atomicAdd(ap + 5, A1.y);
    atomicAdd(ap + 6, A1.z); atomicAdd(ap + 7, A1.w);
    __syncthreads();
    atomicAdd(&o[b * CD + t], acc[t]);
}

// ---------------------------------------------------------------------------
// Kernel 5: small dense GEMM  Y = Xeff @ W^T  (16x256 @ 256x256) via
// V_WMMA_F32_16X16X4_F32, fused epilogue.
//   Xeff = x (+ add).  MODE_COPY: out = Xeff (no GEMM).
//   MODE_GATE: g = sigmoid(Y + bias); out = lastu*g + Xeff*(1-g)
//   MODE_RELU: out = relu(Y + bias)
// 512 threads = 16 waves; wave w owns N-columns [16w, 16w+16).
// ---------------------------------------------------------------------------
#define XLDS_STRIDE 260   // pad so 16 A-rows land on distinct bank groups

__global__ void gemm_epi_kernel(const float* __restrict__ x,
                                const float* __restrict__ add,
                                const float* __restrict__ W,
                                const float* __restrict__ bias,
                                const float* __restrict__ lastu,
                                float* __restrict__ out,
                                int mode)
{
    __shared__ float xs[16 * XLDS_STRIDE];
    int t = threadIdx.x;
    for (int i = t; i < CBD; i += 512) {
        int bb = i >> 8, d = i & 255;
        float v = x[i] + (add ? add[i] : 0.f);
        xs[bb * XLDS_STRIDE + d] = v;
        if (mode == MODE_COPY) out[i] = v;
    }
    __syncthreads();
    if (mode == MODE_COPY) return;

    int wid   = t >> 5;
    int lane  = t & 31;
    int nbase = wid * 16;
    int arow  = lane & 15;          // A: M = lane%16
    int khalf = (lane >> 4) * 2;    // A: VGPR0 holds K=0 (lo half) / K=2 (hi half)

    v8f c = {0.f, 0.f, 0.f, 0.f, 0.f, 0.f, 0.f, 0.f};
    for (int k0 = 0; k0 < CD; k0 += 4) {
        v2f a, bf;
        a.x = xs[arow * XLDS_STRIDE + k0 + khalf];
        a.y = xs[arow * XLDS_STRIDE + k0 + khalf + 1];
        // B[k][n] = W[nbase + n][k]; lane's n = lane%16, k split like A
        const float* wr = W + (long long)(nbase + arow) * CD + k0 + khalf;
        bf.x = wr[0];
        bf.y = wr[1];
        c = __builtin_amdgcn_wmma_f32_16x16x4_f32(
                false, a, false, bf, (short)0, c, false, false);
    }

    // C/D 16x16 f32 layout: VGPR v, lanes 0-15 -> M=v, lanes 16-31 -> M=v+8; N=lane%16
    int n    = nbase + (lane & 15);
    int half = (lane >> 4) * 8;
    float bs = bias[n];
#pragma unroll
    for (int v = 0; v < 8; ++v) {
        int bb = v + half;
        float y  = c[v] + bs;
        float xv = xs[bb * XLDS_STRIDE + n];
        float r;
        if (mode == MODE_GATE) {
            float g = 1.0f / (1.0f + __expf(-y));
            r = lastu[bb * CD + n] * g + xv * (1.0f - g);
        } else {
            r = fmaxf(y, 0.0f);
        }
        out[bb * CD + n] = r;
    }
}

// ---------------------------------------------------------------------------
// Host orchestration
// ---------------------------------------------------------------------------
extern "C" void kernel_launch(void* const* d_in, const int* in_sizes, int n_in,
                              void* d_out, int out_size, void* d_ws, size_t ws_size,
                              hipStream_t stream)
{
    const int*   story     = (const int*)  d_in[0];
    const int*   kb_len    = (const int*)  d_in[1];
    const int*   conv_len  = (const int*)  d_in[2];
    const float* query     = (const float*)d_in[3];   // (B,1,D)
    const float* hist      = (const float*)d_in[4];   // (B,CONV,D)
    const float* query_vec = (const float*)d_in[5];   // (B,D)
    const float* gp        = (const float*)d_in[6];   // (B,M)
    const float* C_emb     = (const float*)d_in[7];   // (H+1,V,D)
    const float* Tg_w      = (const float*)d_in[8];   // (H+1,D,D)
    const float* Tg_b      = (const float*)d_in[9];   // (H+1,D)
    const float* FW_w      = (const float*)d_in[10];  // (D,D)
    const float* FW_b      = (const float*)d_in[11];  // (D,)

    // workspace layout (floats)
    float* ws    = (float*)d_ws;
    float* E     = ws;                      // 3 * BMD  (E[0..2])
    float* s     = ws + 3ll * CBMD;         // B*M logits scratch
    float* stats = s + CBM;                 // B*2 (max, sum)
    float* o     = stats + 2 * CB;          // B*D accumulator
    float* u_a   = o + CBD;                 // 4*B*D
    float* u_b   = u_a + 4 * CBD;           // 4*B*D
    float* qbuf  = u_b + 4 * CBD;           // B*D

    // output layout: psoft | logits | uf | m_story[-1]
    float* outp       = (float*)d_out;
    float* psoft_out  = outp;
    float* logits_out = outp + CBM;
    float* uf         = outp + 2 * CBM;
    float* E3         = outp + 2 * CBM + CBD;

    // 1) materialize all four embedded memories (E3 straight into d_out)
    embed_kernel<<<(4 * CBM) / 8, 256, 0, stream>>>(story, kb_len, conv_len,
                                                    hist, C_emb, E, E3);

    // 2) q = query[:,0,:]
    gemm_epi_kernel<<<1, 512, 0, stream>>>(query, nullptr, nullptr, nullptr,
                                           nullptr, qbuf, MODE_COPY);

    for (int layer = 0; layer < CNLAYER; ++layer) {
        float* cur  = (layer & 1) ? u_b : u_a;
        float* last = (layer & 1) ? u_a : u_b;

        if (layer == 0)
            gemm_epi_kernel<<<1, 512, 0, stream>>>(qbuf, nullptr, nullptr, nullptr,
                                                   nullptr, cur, MODE_COPY);
        else
            gemm_epi_kernel<<<1, 512, 0, stream>>>(qbuf, nullptr, Tg_w, Tg_b,
                                                   last, cur, MODE_GATE);

        for (int hop = 0; hop < CHOPS; ++hop) {
            const float* Eh   = E + (long long)hop * CBMD;
            const float* Ehp1 = (hop + 1 < 3) ? (E + (long long)(hop + 1) * CBMD) : E3;

            logits_kernel<<<CBM / 8, 256, 0, stream>>>(Eh, cur + hop * CBD, nullptr, s);
            softmax_stats_kernel<<<CB, 256, 0, stream>>>(s, stats, o);
            weighted_kernel<<<CB * 16, 256, 0, stream>>>(Ehp1, s, stats, nullptr, o, nullptr);

            if (layer == 0)
                gemm_epi_kernel<<<1, 512, 0, stream>>>(cur + hop * CBD, o,
                    nullptr, nullptr, nullptr, cur + (hop + 1) * CBD, MODE_COPY);
            else
                gemm_epi_kernel<<<1, 512, 0, stream>>>(cur + hop * CBD, o,
                    Tg_w + (long long)(hop + 1) * CD * CD, Tg_b + (hop + 1) * CD,
                    last + (hop + 1) * CBD, cur + (hop + 1) * CBD, MODE_GATE);
        }
        // q = relu(u[3] @ FW^T + FW_b)
        gemm_epi_kernel<<<1, 512, 0, stream>>>(cur + 3 * CBD, nullptr, FW_w, FW_b,
                                               nullptr, qbuf, MODE_RELU);
    }

    // 3) global-pointer phase: uf = query_vector; 3 hops on gp-scaled memories
    gemm_epi_kernel<<<1, 512, 0, stream>>>(query_vec, nullptr, nullptr, nullptr,
                                           nullptr, uf, MODE_COPY);
    for (int hop = 0; hop < CHOPS; ++hop) {
        const float* Eh   = E + (long long)hop * CBMD;
        const float* Ehp1 = (hop + 1 < 3) ? (E + (long long)(hop + 1) * CBMD) : E3;
        float* starget    = (hop == 2) ? logits_out : s;

        logits_kernel<<<CBM / 8, 256, 0, stream>>>(Eh, uf, gp, starget);
        softmax_stats_kernel<<<CB, 256, 0, stream>>>(starget, stats, o);
        weighted_kernel<<<CB * 16, 256, 0, stream>>>(Ehp1, starget, stats, gp, o,
                                                     (hop == 2) ? psoft_out : nullptr);
        gemm_epi_kernel<<<1, 512, 0, stream>>>(uf, o, nullptr, nullptr, nullptr,
                                               uf, MODE_COPY);
    }
}